// HypergraphsBlock_68143951118559
// MI455X (gfx1250) — compile-verified
//
#include <hip/hip_runtime.h>
#include <hip/hip_bf16.h>

typedef float v2f __attribute__((ext_vector_type(2)));
typedef float v8f __attribute__((ext_vector_type(8)));

static constexpr int N_NODES = 50000;
static constexpr int N_EDGES = 20000;
static constexpr int NNZ     = 800000;
static constexpr int F       = 256;       // all feature dims are 256
static constexpr float EPS   = 1e-5f;

// GEMM blocking: 5 row-tiles (80 rows) per block strip, 50000 = 625 * 80
static constexpr int RTILES  = 5;
static constexpr int ROWS_PB = RTILES * 16;   // 80

// ---------------------------------------------------------------------------
// Utility: zero a float buffer
// ---------------------------------------------------------------------------
__global__ void zero_f(float* __restrict__ p, int n) {
    int i = blockIdx.x * 256 + threadIdx.x;
    if (i < n) p[i] = 0.0f;
}

// ---------------------------------------------------------------------------
// Degree (D) and hyperedge cardinality (B) accumulation
// D[node] += w_e[edge]; Bc[edge] += 1
// ---------------------------------------------------------------------------
__global__ void deg_kernel(const int* __restrict__ nidx, const int* __restrict__ eidx,
                           const float* __restrict__ w_e,
                           float* __restrict__ D, float* __restrict__ Bc) {
    int i = blockIdx.x * 256 + threadIdx.x;
    if (i >= NNZ) return;
    int e = eidx[i];
    atomicAdd(&D[nidx[i]], w_e[e]);
    atomicAdd(&Bc[e], 1.0f);
}

__global__ void invert_kernel(float* __restrict__ p, int n) {
    int i = blockIdx.x * 256 + threadIdx.x;
    if (i < n) {
        float v = p[i];
        p[i] = (v != 0.0f) ? 1.0f / v : 0.0f;
    }
}

// ---------------------------------------------------------------------------
// GEMM: Y[N,256] = X[N,256] @ W[256,256] via V_WMMA_F32_16X16X4_F32
// grid = (2, 625); block = 256 (8 wave32 waves).
// Block covers 80 rows x 128 cols. Wave w owns col tile [bx*128 + w*16, +16)
// and all 5 row tiles of the strip (B fragments amortized 5x).
// A panel (80x256 f32 = 80 KB) staged in LDS once per block.
// Inner k-step: 5x ds_load_b64 (A) + 2x global_load_b32 (B) + 5x WMMA.
// ---------------------------------------------------------------------------
__global__ __launch_bounds__(256) void gemm_wmma_f32(
    const float* __restrict__ X, const float* __restrict__ W,
    float* __restrict__ Y)
{
    __shared__ float As[ROWS_PB * 256];   // 80 KB

    const int rowBase = blockIdx.y * ROWS_PB;
    const int tid     = threadIdx.x;

    // Cooperative coalesced load of the 80x256 A panel (float4 = b128 loads)
    {
        const float4* src = (const float4*)(X + (size_t)rowBase * F);
        float4* dst = (float4*)As;
        #pragma unroll
        for (int i = tid; i < ROWS_PB * 256 / 4; i += 256) dst[i] = src[i];
    }
    __syncthreads();

    const int lane = tid & 31;
    const int wave = tid >> 5;
    const int col  = blockIdx.x * 128 + wave * 16 + (lane & 15);
    const int kOff = (lane < 16) ? 0 : 2;   // lanes 0-15: K={0,1}; 16-31: K={2,3}
    const int arow = lane & 15;             // M index (within tile) held by this lane

    v8f acc[RTILES];
    #pragma unroll
    for (int t = 0; t < RTILES; ++t) acc[t] = (v8f){};

    #pragma unroll 2
    for (int kk = 0; kk < 256; kk += 4) {
        // B fragment (shared by all row tiles): W[kk+kOff][col], W[kk+kOff+1][col]
        v2f b;
        b.x = W[(size_t)(kk + kOff) * F + col];
        b.y = W[(size_t)(kk + kOff + 1) * F + col];
        #pragma unroll
        for (int t = 0; t < RTILES; ++t) {
            // A fragment: A[M = t*16+arow][kk+kOff .. +1] (contiguous in LDS)
            v2f a = *(const v2f*)&As[(t * 16 + arow) * 256 + kk + kOff];
            acc[t] = __builtin_amdgcn_wmma_f32_16x16x4_f32(
                /*neg_a=*/false, a, /*neg_b=*/false, b,
                /*c_mod=*/(short)0, acc[t], /*reuse_a=*/false, /*reuse_b=*/false);
        }
    }

    // C layout: VGPR r -> row (r, lanes 0-15) / (r+8, lanes 16-31), N = lane&15
    const int rAdd = (lane >= 16) ? 8 : 0;
    #pragma unroll
    for (int t = 0; t < RTILES; ++t) {
        #pragma unroll
        for (int r = 0; r < 8; ++r) {
            int row = rowBase + t * 16 + r + rAdd;
            Y[(size_t)row * F + col] = acc[t][r];
        }
    }
}

// ---------------------------------------------------------------------------
// Node -> edge scatter: Eacc[edge][f] += Y[node][f]   (one block per nnz)
// ---------------------------------------------------------------------------
__global__ __launch_bounds__(256) void scatter_n2e(
    const float* __restrict__ Y, const int* __restrict__ nidx,
    const int* __restrict__ eidx, float* __restrict__ Eacc)
{
    int nz = blockIdx.x;
    int f  = threadIdx.x;
    int n  = nidx[nz];
    int e  = eidx[nz];
    atomicAdd(&Eacc[(size_t)e * F + f], Y[(size_t)n * F + f]);
}

// ---------------------------------------------------------------------------
// Edge -> node scatter with B^{-1} scaling: S[node][f] += Binv[edge]*E[edge][f]
// ---------------------------------------------------------------------------
__global__ __launch_bounds__(256) void scatter_e2n(
    const float* __restrict__ Eacc, const float* __restrict__ Binv,
    const int* __restrict__ nidx, const int* __restrict__ eidx,
    float* __restrict__ S)
{
    int nz = blockIdx.x;
    int f  = threadIdx.x;
    int n  = nidx[nz];
    int e  = eidx[nz];
    atomicAdd(&S[(size_t)n * F + f], Binv[e] * Eacc[(size_t)e * F + f]);
}

// ---------------------------------------------------------------------------
// Column stats for BatchNorm over v = Dinv[n]*S[n][f] + bias[f]
// One thread per feature per block; each block covers 128 rows; partials
// merged via global float atomics (391 * 256 atomics total -> negligible).
// ---------------------------------------------------------------------------
__global__ __launch_bounds__(256) void colstats(
    const float* __restrict__ S, const float* __restrict__ Dinv,
    const float* __restrict__ bias,
    float* __restrict__ sum, float* __restrict__ sumsq)
{
    int f  = threadIdx.x;
    int r0 = blockIdx.x * 128;
    int r1 = r0 + 128; if (r1 > N_NODES) r1 = N_NODES;
    float bf = bias[f];
    float s = 0.0f, s2 = 0.0f;
    for (int r = r0; r < r1; ++r) {
        float v = Dinv[r] * S[(size_t)r * F + f] + bf;
        s  += v;
        s2 += v * v;
    }
    atomicAdd(&sum[f], s);
    atomicAdd(&sumsq[f], s2);
}

// ---------------------------------------------------------------------------
// Fused BN (train-mode, biased var) + ReLU
// out[n][f] = relu((v - mu) * rsqrt(var+eps) * g[f] + be[f]),
//   v = Dinv[n]*S[n][f] + bias[f]
// ---------------------------------------------------------------------------
__global__ __launch_bounds__(256) void bn_relu(
    const float* __restrict__ S, const float* __restrict__ Dinv,
    const float* __restrict__ bias,
    const float* __restrict__ sum, const float* __restrict__ sumsq,
    const float* __restrict__ g, const float* __restrict__ be,
    float* __restrict__ out)
{
    int i = blockIdx.x * 256 + threadIdx.x;
    if (i >= N_NODES * F) return;
    int f = i & (F - 1);
    int n = i >> 8;
    const float invN = 1.0f / (float)N_NODES;
    float mu  = sum[f] * invN;
    float var = sumsq[f] * invN - mu * mu;
    float v   = Dinv[n] * S[i] + bias[f];
    float o   = (v - mu) * rsqrtf(var + EPS) * g[f] + be[f];
    out[i] = fmaxf(o, 0.0f);
}

// ---------------------------------------------------------------------------
// One HypergraphLayer
// ---------------------------------------------------------------------------
static void run_layer(const float* xin, const float* W, const float* b,
                      const float* g, const float* be,
                      const int* nidx, const int* eidx,
                      float* Dinv, float* Binv, float* sum, float* sumsq,
                      float* Ybuf /*N*F, reused as S*/, float* Ebuf /*E*F*/,
                      float* out, hipStream_t stream)
{
    // y = x @ W
    gemm_wmma_f32<<<dim3(2, N_NODES / ROWS_PB), 256, 0, stream>>>(xin, W, Ybuf);
    // e_sum = segsum(y[node] by edge)
    zero_f<<<(N_EDGES * F + 255) / 256, 256, 0, stream>>>(Ebuf, N_EDGES * F);
    scatter_n2e<<<NNZ, 256, 0, stream>>>(Ybuf, nidx, eidx, Ebuf);
    // s = segsum(Binv[edge]*e_sum[edge] by node)   (Ybuf reused as S)
    zero_f<<<(N_NODES * F + 255) / 256, 256, 0, stream>>>(Ybuf, N_NODES * F);
    scatter_e2n<<<NNZ, 256, 0, stream>>>(Ebuf, Binv, nidx, eidx, Ybuf);
    // BN stats over v = Dinv*s + b
    zero_f<<<2, 256, 0, stream>>>(sum, 2 * F);   // sum & sumsq contiguous
    colstats<<<(N_NODES + 127) / 128, 256, 0, stream>>>(Ybuf, Dinv, b, sum, sumsq);
    // normalize + relu
    bn_relu<<<(N_NODES * F + 255) / 256, 256, 0, stream>>>(
        Ybuf, Dinv, b, sum, sumsq, g, be, out);
}

// ---------------------------------------------------------------------------
extern "C" void kernel_launch(void* const* d_in, const int* in_sizes, int n_in,
                              void* d_out, int out_size, void* d_ws, size_t ws_size,
                              hipStream_t stream) {
    const float* x    = (const float*)d_in[0];
    const int*   hidx = (const int*)  d_in[1];   // [2, NNZ]
    const float* w_e  = (const float*)d_in[2];
    const float* W1   = (const float*)d_in[3];
    const float* b1   = (const float*)d_in[4];
    const float* g1   = (const float*)d_in[5];
    const float* be1  = (const float*)d_in[6];
    const float* W2   = (const float*)d_in[7];
    const float* b2   = (const float*)d_in[8];
    const float* g2   = (const float*)d_in[9];
    const float* be2  = (const float*)d_in[10];

    const int* nidx = hidx;         // row 0
    const int* eidx = hidx + NNZ;   // row 1

    // Workspace layout (floats)
    float* ws    = (float*)d_ws;
    float* Dinv  = ws;                         // N_NODES
    float* Binv  = Dinv + N_NODES;             // N_EDGES
    float* sum   = Binv + N_EDGES;             // F
    float* sumsq = sum + F;                    // F
    float* Ybuf  = sumsq + F;                  // N_NODES * F  (y, then s)
    float* Ebuf  = Ybuf + (size_t)N_NODES * F; // N_EDGES * F
    float* Hbuf  = Ebuf + (size_t)N_EDGES * F; // N_NODES * F  (layer-1 output)

    // Degrees / cardinalities (shared by both layers)
    zero_f<<<(N_NODES + N_EDGES + 255) / 256, 256, 0, stream>>>(Dinv, N_NODES + N_EDGES);
    deg_kernel<<<(NNZ + 255) / 256, 256, 0, stream>>>(nidx, eidx, w_e, Dinv, Binv);
    invert_kernel<<<(N_NODES + N_EDGES + 255) / 256, 256, 0, stream>>>(Dinv, N_NODES + N_EDGES);

    // Layer 1: x -> Hbuf
    run_layer(x, W1, b1, g1, be1, nidx, eidx,
              Dinv, Binv, sum, sumsq, Ybuf, Ebuf, Hbuf, stream);
    // Layer 2: Hbuf -> d_out
    run_layer(Hbuf, W2, b2, g2, be2, nidx, eidx,
              Dinv, Binv, sum, sumsq, Ybuf, Ebuf, (float*)d_out, stream);
}